// Upsample2x2x2_40724879901079
// MI455X (gfx1250) — compile-verified
//
#include <hip/hip_runtime.h>

// 2x2x2 upsample via lhs-dilated conv with separable binomial [1,4,6,4,1]/8,
// edge-clamped. Per axis:
//   out[2m]   = (x[m-1] + 6 x[m] + x[m+1]) / 8
//   out[2m+1] = (x[m] + x[m+1]) / 2
// Memory-bound (~254 MB traffic -> ~11 us @ 23.3 TB/s). Uses CDNA5 async
// global->LDS DMA for halo tile staging + B128 output stores.

namespace {

constexpr int IN_D   = 48;           // input D = H = W
constexpr int OUT_D  = 96;
constexpr int NC     = 64;           // N*C = 2*32
constexpr int TILE   = 8;            // input tile edge -> 16^3 output tile
constexpr int TPD    = IN_D / TILE;  // 6 tiles per axis
constexpr int LDS_D  = TILE + 2;     // 10 (1-halo in d and h)
constexpr int LDS_W  = TILE + 2;     // 10 valid w entries (lw = gw - w0 + 1)
constexpr int LDS_WP = 12;           // padded row stride (words)
constexpr int LDS_ELEMS = LDS_D * LDS_D * LDS_W;  // 1000

__device__ __forceinline__ int clampi(int v, int hi) {
  v = v < 0 ? 0 : v;
  return v > hi ? hi : v;
}

// Async DMA one dword from global to LDS (per-lane gather addresses).
__device__ __forceinline__ void async_g2l_b32(const float* gsrc, float* ldst) {
#if defined(__gfx1250__) && __has_builtin(__builtin_amdgcn_global_load_async_to_lds_b32)
  __builtin_amdgcn_global_load_async_to_lds_b32(
      (__attribute__((address_space(1))) int*)(unsigned long long)(const void*)gsrc,
      (__attribute__((address_space(3))) int*)(unsigned)(unsigned long long)(void*)ldst,
      0, 0);
#elif defined(__gfx1250__)
  unsigned lds_off = (unsigned)(unsigned long long)(void*)ldst;  // flat addr[31:0] == LDS offset
  asm volatile("global_load_async_to_lds_b32 %0, %1, off"
               :: "v"(lds_off), "v"(gsrc)
               : "memory");
#else
  *ldst = *gsrc;  // host-pass / non-gfx1250 parse path
#endif
}

__device__ __forceinline__ void wait_async_zero() {
#if defined(__gfx1250__) && __has_builtin(__builtin_amdgcn_s_wait_asynccnt)
  __builtin_amdgcn_s_wait_asynccnt(0);
#elif defined(__gfx1250__)
  asm volatile("s_wait_asynccnt 0" ::: "memory");
#endif
}

}  // namespace

__global__ __launch_bounds__(256)
void upsample_blur_kernel(const float* __restrict__ x, float* __restrict__ out) {
  __shared__ float smem[LDS_D * LDS_D * LDS_WP];

  const int tid = threadIdx.x;
  const int ch  = blockIdx.y;

  const int tIdx = blockIdx.x;
  const int tw = tIdx % TPD;
  const int th = (tIdx / TPD) % TPD;
  const int td = tIdx / (TPD * TPD);

  const int d0 = td * TILE, h0 = th * TILE, w0t = tw * TILE;

  const float* xc = x + (size_t)ch * (IN_D * IN_D * IN_D);

  // ---- Stage clamped-halo 10x10x10 input tile into LDS via async DMA ----
  for (int idx = tid; idx < LDS_ELEMS; idx += 256) {
    const int lw = idx % LDS_W;
    const int t2 = idx / LDS_W;
    const int lh = t2 % LDS_D;
    const int ld = t2 / LDS_D;
    const int gd = clampi(d0  + ld - 1, IN_D - 1);
    const int gh = clampi(h0  + lh - 1, IN_D - 1);
    const int gw = clampi(w0t + lw - 1, IN_D - 1);
    async_g2l_b32(xc + ((gd * IN_D + gh) * IN_D + gw),
                  &smem[(ld * LDS_D + lh) * LDS_WP + lw]);
  }
  wait_async_zero();
  __syncthreads();

  // ---- Each thread: 3x3x4 neighborhood -> 2x2x4 outputs ----
  const int wp = tid & 3;          // w-pair 0..3
  const int hh = (tid >> 2) & 7;   // h 0..7
  const int mm = tid >> 5;         // d 0..7
  const int w0 = wp * 2;           // LDS w base (covers lw w0..w0+3)

  // D-combine: even = a + 6b + c, odd = b + c  (scaling deferred)
  float Ed[3][4], Odp[3][4];
#pragma unroll
  for (int j = 0; j < 3; ++j) {
#pragma unroll
    for (int k = 0; k < 4; ++k) {
      const int base = (hh + j) * LDS_WP + (w0 + k);
      const float a = smem[(mm + 0) * (LDS_D * LDS_WP) + base];
      const float b = smem[(mm + 1) * (LDS_D * LDS_WP) + base];
      const float c = smem[(mm + 2) * (LDS_D * LDS_WP) + base];
      Ed[j][k]  = fmaf(6.0f, b, a + c);
      Odp[j][k] = b + c;
    }
  }

  // H-combine
  float Gee[4], Geo[4], Goe[4], Goo[4];
#pragma unroll
  for (int k = 0; k < 4; ++k) {
    Gee[k] = fmaf(6.0f, Ed[1][k],  Ed[0][k]  + Ed[2][k]);
    Geo[k] = Ed[1][k]  + Ed[2][k];
    Goe[k] = fmaf(6.0f, Odp[1][k], Odp[0][k] + Odp[2][k]);
    Goo[k] = Odp[1][k] + Odp[2][k];
  }

  // W-combine + scaled B128 stores. Scale = (1/8)^3 * 4^(#odd axes).
  const size_t obase = (size_t)ch * ((size_t)OUT_D * OUT_D * OUT_D);
  const int ow  = 2 * (w0t + w0);     // multiple of 4 -> 16B aligned
  const int ode = 2 * (d0 + mm);
  const int ohe = 2 * (h0 + hh);

  auto emit = [&](const float* G, int od, int oh, float s) {
    float4 v;
    v.x = fmaf(6.0f, G[1], G[0] + G[2]) * s;          // even w
    v.y = (G[1] + G[2]) * (4.0f * s);                 // odd  w
    v.z = fmaf(6.0f, G[2], G[1] + G[3]) * s;          // even w
    v.w = (G[2] + G[3]) * (4.0f * s);                 // odd  w
    *reinterpret_cast<float4*>(out + obase + ((size_t)od * OUT_D + oh) * OUT_D + ow) = v;
  };

  constexpr float S = 1.0f / 512.0f;
  emit(Gee, ode,     ohe,     S);
  emit(Geo, ode,     ohe + 1, 4.0f * S);
  emit(Goe, ode + 1, ohe,     4.0f * S);
  emit(Goo, ode + 1, ohe + 1, 16.0f * S);
}

extern "C" void kernel_launch(void* const* d_in, const int* in_sizes, int n_in,
                              void* d_out, int out_size, void* d_ws, size_t ws_size,
                              hipStream_t stream) {
  (void)in_sizes; (void)n_in; (void)d_ws; (void)ws_size; (void)out_size;
  const float* x = (const float*)d_in[0];   // (2,32,48,48,48) fp32
  float* out = (float*)d_out;               // (2,32,96,96,96) fp32

  dim3 grid(TPD * TPD * TPD, NC);           // 216 tiles x 64 channels
  upsample_blur_kernel<<<grid, dim3(256), 0, stream>>>(x, out);
}